// TopDownPropagate_50216757624910
// MI455X (gfx1250) — compile-verified
//
#include <hip/hip_runtime.h>

// TopDownPropagate for MI455X (gfx1250, wave32).
// One 32-thread workgroup per batch; ent_attn[b] (32x128 f32 = 16KB) lives in
// LDS for all 16 tree levels; cross-lane dot-product reductions are done with
// V_WMMA_F32_16X16X4_F32 (ones-matrix row-sum trick) + one shfl_xor(16).

typedef float v2f __attribute__((ext_vector_type(2)));
typedef float v8f __attribute__((ext_vector_type(8)));

#define BSZ   1024
#define NSEQ  32
#define NBOX  128
#define NCTX  5
#define MAXC  4

// Full 32-lane f32 sum, result valid in every lane.
// A-matrix 16x4 f32 layout (ISA 7.12.2): lanes 0-15 VGPR0->K=0, VGPR1->K=1;
// lanes 16-31 VGPR0->K=2, VGPR1->K=3; M = lane%16.
// With A = {v, 0} and B = ones: D[m,j] = v[m] + v[m+16].
// D VGPR r: lanes 0-15 hold (M=r), lanes 16-31 hold (M=8+r) => summing the 8
// D regs per lane gives sum(u[0..7]) in low half, sum(u[8..15]) in high half.
__device__ __forceinline__ float wave_sum32(float v) {
    v2f a; a[0] = v;    a[1] = 0.0f;
    v2f b; b[0] = 1.0f; b[1] = 1.0f;
    v8f c = {};
    v8f d = __builtin_amdgcn_wmma_f32_16x16x4_f32(
        /*neg_a=*/false, a, /*neg_b=*/false, b,
        /*c_mod=*/(short)0, c, /*reuse_a=*/false, /*reuse_b=*/false);
    float s = d[0] + d[1] + d[2] + d[3] + d[4] + d[5] + d[6] + d[7];
    s += __shfl_xor(s, 16, 32);
    return s;
}

__global__ __launch_bounds__(32)
void TopDownPropagate_50216757624910_kernel(
    const float* __restrict__ ent_attn,      // (BSZ, NSEQ, NBOX)
    const float* __restrict__ spo_attn,      // (BSZ, NSEQ, NBOX, NCTX)
    const int*   __restrict__ ctx_idx,       // (BSZ, NSEQ, NBOX, NCTX)
    const float* __restrict__ roi_mask,      // (BSZ, NSEQ, NBOX, NCTX)
    const int*   __restrict__ roi_cls,       // (BSZ, NBOX)
    const float* __restrict__ w_child,       // (BSZ, NSEQ, NBOX)
    const int*   __restrict__ parent_idx,    // (L, BSZ)
    const float* __restrict__ parent_valid,  // (L, BSZ)
    const int*   __restrict__ child_idx,     // (L, BSZ, MAXC)
    const float* __restrict__ child_valid,   // (L, BSZ, MAXC)
    const int*   __restrict__ edge_idx,      // (L, BSZ, MAXC)
    const int*   __restrict__ flat_sample,   // (L, BSZ)
    const int*   __restrict__ flat_slot,     // (L, BSZ)
    float*       __restrict__ out,           // (BSZ, NSEQ, NBOX)
    int L)
{
    __shared__ float ent[NSEQ * NBOX];       // 16 KB: this batch's slab

    const int b    = blockIdx.x;
    const int lane = threadIdx.x;            // 0..31, wave32

    // Bring ent_attn[b] into LDS (coalesced: 128 iterations of 32 lanes).
    for (int i = lane; i < NSEQ * NBOX; i += 32)
        ent[i] = ent_attn[(size_t)b * (NSEQ * NBOX) + i];
    __syncthreads();

    // Per-lane box ownership: boxes n = lane + 32*q, q=0..3.
    int   rcls[4];
    float cls[4];
#pragma unroll
    for (int q = 0; q < 4; ++q) {
        const int n = lane + 32 * q;
        rcls[q] = roi_cls[b * NBOX + n];
        cls[q]  = (rcls[q] != -1) ? 1.0f : 0.0f;
    }

    for (int it = 0; it < L; ++it) {
        const int   base = it * BSZ + b;
        const int   pidx = parent_idx[base];
        const float pval = parent_valid[base];
        const int   c0   = child_idx[base * MAXC + 0];   // only slot 0 feeds output
        const float cv0  = child_valid[base * MAXC + 0];
        const int   e0   = edge_idx[base * MAXC + 0];
        const int   fs   = flat_sample[base];
        const int   fl   = flat_slot[base];

        // transfer[c] = sum_n  (ent[c0][n]*cv0*cls[n]) *
        //               (spo[b,e0,n,c]*roi*cv0*cls[n]) * roi   + 1e-6
        float partial[NCTX];
#pragma unroll
        for (int c = 0; c < NCTX; ++c) partial[c] = 0.0f;

        const size_t row_base = (((size_t)b * NSEQ + (size_t)e0) * NBOX) * NCTX;
#pragma unroll
        for (int q = 0; q < 4; ++q) {
            const int    n   = lane + 32 * q;
            const float  a_n = ent[c0 * NBOX + n] * cv0 * cls[q];
            const float* sp  = spo_attn + row_base + (size_t)n * NCTX;
            const float* rm  = roi_mask + row_base + (size_t)n * NCTX;
#pragma unroll
            for (int c = 0; c < NCTX; ++c) {
                const float r  = rm[c];
                const float ks = sp[c] * r * cv0 * cls[q];
                partial[c] += a_n * ks * r;
            }
        }

        float transfer[NCTX];
#pragma unroll
        for (int c = 0; c < NCTX; ++c)
            transfer[c] = wave_sum32(partial[c]) + 1e-6f;   // v_wmma_f32_16x16x4_f32

        // Scatter columns: ctx_idx[fs, edge_idx[it,fs,fl], 0, :]  (lane-uniform)
        const int e_fs = edge_idx[(it * BSZ + fs) * MAXC + fl];
        int ucols[NCTX];
#pragma unroll
        for (int c = 0; c < NCTX; ++c)
            ucols[c] = ctx_idx[(((size_t)fs * NSEQ + (size_t)e_fs) * NBOX) * NCTX + c];

        // Parent row update + row-max for normalization.
        const size_t wrow = ((size_t)b * NSEQ + (size_t)pidx) * NBOX;
        float upd[4], oldv[4];
        float mymax = 0.0f;
#pragma unroll
        for (int q = 0; q < 4; ++q) {
            const int n = lane + 32 * q;
            float ats = 1e-6f;
#pragma unroll
            for (int c = 0; c < NCTX; ++c)
                ats = (n == ucols[c]) ? transfer[c] : ats;
            oldv[q] = ent[pidx * NBOX + n];
            const float u = oldv[q] + ats * w_child[wrow + n];
            upd[q] = u;
            mymax  = fmaxf(mymax, fabsf(u));
        }
        // max over all 32 lanes (butterfly, no divergence)
#pragma unroll
        for (int off = 1; off < 32; off <<= 1)
            mymax = fmaxf(mymax, __shfl_xor(mymax, off, 32));
        const float denom = (mymax <= 1.0f) ? 1.0f : mymax;

#pragma unroll
        for (int q = 0; q < 4; ++q) {
            const int n = lane + 32 * q;
            float u = upd[q] / denom;
            u = (rcls[q] == -1) ? -1.0f : u;
            const float nv = (pval > 0.0f) ? u : oldv[q];
            ent[pidx * NBOX + n] = nv;
        }
        __syncthreads();   // make this level's row writes visible to next level
    }

    for (int i = lane; i < NSEQ * NBOX; i += 32)
        out[(size_t)b * (NSEQ * NBOX) + i] = ent[i];
}

extern "C" void kernel_launch(void* const* d_in, const int* in_sizes, int n_in,
                              void* d_out, int out_size, void* d_ws, size_t ws_size,
                              hipStream_t stream) {
    (void)n_in; (void)out_size; (void)d_ws; (void)ws_size;
    const float* ent_attn     = (const float*)d_in[0];
    const float* spo_attn     = (const float*)d_in[1];
    const int*   ctx_idx      = (const int*)  d_in[2];
    const float* roi_mask     = (const float*)d_in[3];
    const int*   roi_cls      = (const int*)  d_in[4];
    const float* w_child      = (const float*)d_in[5];
    const int*   parent_idx   = (const int*)  d_in[6];
    const float* parent_valid = (const float*)d_in[7];
    const int*   child_idx    = (const int*)  d_in[8];
    const float* child_valid  = (const float*)d_in[9];
    const int*   edge_idx     = (const int*)  d_in[10];
    const int*   flat_sample  = (const int*)  d_in[11];
    const int*   flat_slot    = (const int*)  d_in[12];

    const int L = in_sizes[6] / BSZ;   // parent_idx is (L, BSZ)

    TopDownPropagate_50216757624910_kernel<<<BSZ, 32, 0, stream>>>(
        ent_attn, spo_attn, ctx_idx, roi_mask, roi_cls, w_child,
        parent_idx, parent_valid, child_idx, child_valid, edge_idx,
        flat_sample, flat_slot, (float*)d_out, L);
}